// CapsuleLayer_46677704573208
// MI455X (gfx1250) — compile-verified
//
#include <hip/hip_runtime.h>
#include <math.h>

// Problem constants (from reference)
#define NCAPS 10
#define RDIM  1152
#define ICH   8
#define OCH   16
#define BATCH 256
#define KDIM  (RDIM * ICH)          // 9216
#define BR    (BATCH * RDIM)        // 294912
#define CBR   (NCAPS * BR)          // 2949120
#define KS    24                     // split-K factor
#define RCHUNK (RDIM / KS)           // 48 r's per K-chunk

typedef __attribute__((ext_vector_type(2))) float v2f;
typedef __attribute__((ext_vector_type(8))) float v8f;

// ---------------------------------------------------------------------------
// GEMM: out_part[ks][b][c][o] = sum over r in chunk, i of (p[c,b,r]*x[b,r,i]) * W[c,r,i,o]
// One wave = one 16(b) x 16(o) tile for capsule c, K-chunk ks.
// A fragment (16x4 f32): lane<16 holds rows M=lane, K pair {k,k+1}; lanes>=16 hold {k+2,k+3}.
// B fragment (4x16 f32): mirrored (lane-half selects K pair, lane%16 = N).
// C/D (16x16 f32): VGPR v, lanes0-15 -> M=v, lanes16-31 -> M=v+8, N=lane%16.
// UNIFORM is a template parameter so iteration 0 (p == 0.1 everywhere) gets a
// branch-free inner loop with the scale folded to an inline constant.
// ---------------------------------------------------------------------------
template <bool UNIFORM>
__global__ __launch_bounds__(256) void caps_gemm(
    const float* __restrict__ x,       // [B][R][I]
    const float* __restrict__ W,       // [C][R][I][O]
    const float* __restrict__ p,       // [C][B][R]
    float* __restrict__ out_part)      // [KS][B][C][O]
{
  const int ks   = blockIdx.x;
  const int c    = blockIdx.y;
  const int wave = threadIdx.x >> 5;
  const int lane = threadIdx.x & 31;
  const int half = lane >> 4;          // 0: lanes 0-15, 1: lanes 16-31
  const int l    = lane & 15;
  const int mtile = blockIdx.z * 8 + wave;   // 0..15
  const int b0   = mtile * 16;
  const int bA   = b0 + l;             // batch row this lane feeds into A
  const int r0   = ks * RCHUNK;

  const float* xrow = x + (size_t)bA * KDIM;
  const float* prow = p + ((size_t)c * BATCH + bA) * RDIM;

  v8f acc = {};
  #pragma unroll 2
  for (int r = r0; r < r0 + RCHUNK; ++r) {
    const float pv = UNIFORM ? 0.1f : prow[r];
    const float* wrow = W + (size_t)(c * RDIM + r) * (ICH * OCH);  // 128 floats
    const float* xk   = xrow + r * ICH;
    #pragma unroll
    for (int s = 0; s < 2; ++s) {                 // two WMMA steps cover i=0..7
      v2f a = *(const v2f*)(xk + 4 * s + 2 * half);
      a.x *= pv;
      a.y *= pv;
      v2f bf;
      bf.x = wrow[(4 * s + 2 * half)     * OCH + l];
      bf.y = wrow[(4 * s + 2 * half + 1) * OCH + l];
      acc = __builtin_amdgcn_wmma_f32_16x16x4_f32(
          /*neg_a=*/false, a, /*neg_b=*/false, bf,
          /*c_mod=*/(short)0, acc, /*reuse_a=*/false, /*reuse_b=*/false);
    }
  }

  float* op = out_part + (size_t)ks * (BATCH * NCAPS * OCH);
  #pragma unroll
  for (int v = 0; v < 8; ++v) {
    const int b = b0 + 8 * half + v;
    op[((size_t)b * NCAPS + c) * OCH + l] = acc[v];
  }
}

// ---------------------------------------------------------------------------
// Reduce split-K partials, squash elementwise (x|x|/(1+x^2)), produce
// S[b,c] = sum_o squash(out). Last iteration writes squashed output to d_out.
// One thread per (b,c).
// ---------------------------------------------------------------------------
__global__ void caps_reduce_squash(const float* __restrict__ out_part,
                                   float* __restrict__ Sbuf,
                                   float* __restrict__ dout,
                                   int last)
{
  const int t = blockIdx.x * blockDim.x + threadIdx.x;   // 0 .. B*C-1
  if (t >= BATCH * NCAPS) return;
  float o16[OCH];
  #pragma unroll
  for (int j = 0; j < OCH; ++j) o16[j] = 0.f;
  for (int ks = 0; ks < KS; ++ks) {
    const float* src = out_part + (size_t)ks * (BATCH * NCAPS * OCH) + (size_t)t * OCH;
    #pragma unroll
    for (int j = 0; j < OCH; ++j) o16[j] += src[j];
  }
  float S = 0.f;
  #pragma unroll
  for (int j = 0; j < OCH; ++j) {
    const float v = o16[j];
    const float y = v * fabsf(v) / (1.f + v * v);   // squash over size-1 axis
    S += y;
    if (last) dout[(size_t)t * OCH + j] = y;
  }
  if (!last) Sbuf[t] = S;
}

// ---------------------------------------------------------------------------
// logits[c][b][r] += p[c][b][r] * S[b][c]   (first iter: logits := 0.1*S)
// ---------------------------------------------------------------------------
__global__ void caps_update(const float* __restrict__ p,
                            const float* __restrict__ Sbuf,
                            float* __restrict__ logits,
                            int first)
{
  const int e = blockIdx.x * blockDim.x + threadIdx.x;   // < C*B*R
  if (e >= CBR) return;
  const int c   = e / BR;
  const int rem = e - c * BR;
  const int b   = rem / RDIM;
  const float pv   = first ? 0.1f : p[e];
  const float base = first ? 0.f  : logits[e];
  logits[e] = base + pv * Sbuf[b * NCAPS + c];
}

// ---------------------------------------------------------------------------
// p[c][b][r] = softmax over c of logits[:, b, r]   (one thread per (b,r))
// ---------------------------------------------------------------------------
__global__ void caps_softmax(const float* __restrict__ logits,
                             float* __restrict__ p)
{
  const int idx = blockIdx.x * blockDim.x + threadIdx.x;   // < B*R
  if (idx >= BR) return;
  float v[NCAPS];
  float m = -3.402823466e+38f;
  #pragma unroll
  for (int c = 0; c < NCAPS; ++c) {
    v[c] = logits[(size_t)c * BR + idx];
    m = fmaxf(m, v[c]);
  }
  float s = 0.f;
  #pragma unroll
  for (int c = 0; c < NCAPS; ++c) {
    const float e = __expf(v[c] - m);
    v[c] = e;
    s += e;
  }
  const float inv = 1.f / s;
  #pragma unroll
  for (int c = 0; c < NCAPS; ++c)
    p[(size_t)c * BR + idx] = v[c] * inv;
}

extern "C" void kernel_launch(void* const* d_in, const int* in_sizes, int n_in,
                              void* d_out, int out_size, void* d_ws, size_t ws_size,
                              hipStream_t stream) {
  (void)in_sizes; (void)n_in; (void)out_size; (void)ws_size;
  const float* x = (const float*)d_in[0];          // [B,R,I]
  const float* W = (const float*)d_in[1];          // [C,R,I,O]
  float* out = (float*)d_out;                      // [B,C,1,1,O] flat

  float* p_buf    = (float*)d_ws;                  // C*B*R
  float* logits   = p_buf + CBR;                   // C*B*R
  float* out_part = logits + CBR;                  // KS*B*C*O
  float* Sbuf     = out_part + (size_t)KS * BATCH * NCAPS * OCH;   // B*C

  const dim3 blk(256);
  const dim3 gGemm(KS, NCAPS, 2);
  const dim3 gRed((BATCH * NCAPS + 255) / 256);
  const dim3 gUpd(CBR / 256);
  const dim3 gSmx(BR / 256);

  // ---- iteration 0 (probs uniform 0.1; softmax of zero logits) ----
  caps_gemm<true><<<gGemm, blk, 0, stream>>>(x, W, p_buf, out_part);
  caps_reduce_squash<<<gRed, blk, 0, stream>>>(out_part, Sbuf, out, /*last=*/0);
  caps_update<<<gUpd, blk, 0, stream>>>(p_buf, Sbuf, logits, /*first=*/1);

  // ---- iteration 1 ----
  caps_softmax<<<gSmx, blk, 0, stream>>>(logits, p_buf);
  caps_gemm<false><<<gGemm, blk, 0, stream>>>(x, W, p_buf, out_part);
  caps_reduce_squash<<<gRed, blk, 0, stream>>>(out_part, Sbuf, out, 0);
  caps_update<<<gUpd, blk, 0, stream>>>(p_buf, Sbuf, logits, 0);

  // ---- iteration 2 (last; logit update is dead code in the reference) ----
  caps_softmax<<<gSmx, blk, 0, stream>>>(logits, p_buf);
  caps_gemm<false><<<gGemm, blk, 0, stream>>>(x, W, p_buf, out_part);
  caps_reduce_squash<<<gRed, blk, 0, stream>>>(out_part, Sbuf, out, /*last=*/1);
}